// Attn_18502719111512
// MI455X (gfx1250) — compile-verified
//
#include <hip/hip_runtime.h>
#include <hip/hip_bf16.h>

// Problem constants (from reference): H = 1024, B = 64, T = 1024.
#define H 1024
#define B 64
#define T 1024
#define TT 64             // t-tile width per workgroup
#define NTILE (T / TT)    // 16 tiles per batch row
#define NTHREADS 256      // 8 waves (wave32)
#define NWAVES 8
#define ROWSTRIDE 65      // padded LDS row stride (floats); TDM pad reproduces this

typedef __attribute__((ext_vector_type(2))) float v2f;
typedef __attribute__((ext_vector_type(8))) float v8f;
typedef unsigned int u32x4 __attribute__((ext_vector_type(4)));
typedef int          i32x4 __attribute__((ext_vector_type(4)));
typedef int          i32x8 __attribute__((ext_vector_type(8)));

// LDS layout (in floats)
#define TILE_OFF   0
#define TILE_FLTS  (H * ROWSTRIDE)            // 66560 (266,240 B)
#define ACC_OFF    (TILE_OFF + TILE_FLTS)     // per-wave accumulators
#define ACC_FLTS   (NWAVES * H)               // 8192
#define DH_OFF     (ACC_OFF + ACC_FLTS)       // decoder_hidden row
#define DH_FLTS    H
#define SC_OFF     (DH_OFF + DH_FLTS)         // 64 scores
#define SC_FLTS    TT
#define SMEM_FLTS  (SC_OFF + SC_FLTS)
#define SMEM_BYTES (SMEM_FLTS * 4)            // 303,360 B <= 320 KB WGP LDS

__global__ __launch_bounds__(NTHREADS, 1)
void attn_tile_kernel(const float* __restrict__ dh,
                      const float* __restrict__ enc,
                      float* __restrict__ partial)
{
    extern __shared__ float lds[];
    float* tile = lds + TILE_OFF;
    float* accw = lds + ACC_OFF;
    float* dlds = lds + DH_OFF;
    float* sc   = lds + SC_OFF;

    const int tid = threadIdx.x;
    const int b   = blockIdx.x >> 4;                 // batch index
    const int t0  = (blockIdx.x & (NTILE - 1)) * TT; // tile start in t

    // Stage decoder_hidden[b, :] and clear per-wave accumulators.
    for (int h = tid; h < H; h += NTHREADS) dlds[h] = dh[b * H + h];
    for (int i = tid; i < ACC_FLTS; i += NTHREADS) accw[i] = 0.0f;

#if __has_builtin(__builtin_amdgcn_tensor_load_to_lds)
    // ---- TDM: DMA the 2D tile enc[0:H, b, t0:t0+TT] straight into LDS ----
    // tile_dim0 = 64 contiguous t's per row; tile_dim1 = 1024 h-rows with
    // stride B*T; pad 1 DWORD after every 64 DWORDs -> ROWSTRIDE=65 layout.
    if (tid < 32) {  // one wave issues the DMA (wave-uniform branch)
        const unsigned long long gaddr =
            (unsigned long long)(uintptr_t)(enc + (size_t)b * T + (size_t)t0);
        const unsigned int laddr = (unsigned int)(uintptr_t)tile; // LDS aperture: addr[31:0] = LDS offset

        u32x4 g0;
        g0[0] = 1u;                                   // count=1 (valid), load, user mode
        g0[1] = laddr;                                // lds_addr (bytes)
        g0[2] = (unsigned int)gaddr;                  // global_addr[31:0]
        g0[3] = (unsigned int)((gaddr >> 32) & 0x1FFFFFFu) | (2u << 30); // addr[56:32] | type=2

        i32x8 g1;
        g1[0] = (int)((2u << 16) | (1u << 20) | (5u << 22)); // data_size=4B, pad_en, pad_interval=64DW, pad=1DW
        g1[1] = (int)(0x400u << 16);                  // tensor_dim0 = 1024 (bits [79:48] low half)
        g1[2] = (int)(0x400u << 16);                  // tensor_dim0 hi=0 | tensor_dim1 = 1024 (low half)
        g1[3] = (int)((unsigned)TT << 16);            // tensor_dim1 hi=0 | tile_dim0 = 64
        g1[4] = (int)TT * 16;                         // tile_dim1 = 1024; tile_dim2 = 0
        g1[5] = (int)(B * T);                         // tensor_dim0_stride = 65536 (low 32 of 48)
        g1[6] = 0;                                    // stride hi, tensor_dim1_stride lo
        g1[7] = 0;

        i32x4 zz = {0, 0, 0, 0};
#if defined(__clang_major__) && __clang_major__ >= 23
        i32x8 z8 = {0, 0, 0, 0, 0, 0, 0, 0};
        __builtin_amdgcn_tensor_load_to_lds(g0, g1, zz, zz, z8, 0);
#else
        __builtin_amdgcn_tensor_load_to_lds(g0, g1, zz, zz, 0);
#endif
        __builtin_amdgcn_s_wait_tensorcnt(0);         // LDS writes complete before barrier signal
    }
#else
    // Fallback: manual staged copy (coalesced float4 along t).
    {
        const int hs = tid >> 4;
        const int t4 = tid & 15;
        const float* gp = enc + (size_t)b * T + (size_t)t0 + (size_t)t4 * 4;
        #pragma unroll 4
        for (int r = 0; r < H / 16; ++r) {
            const int h = r * 16 + hs;
            const float4 v = *(const float4*)(gp + (size_t)h * (B * T));
            float* dp = tile + h * ROWSTRIDE + t4 * 4;
            dp[0] = v.x; dp[1] = v.y; dp[2] = v.z; dp[3] = v.w;
        }
    }
#endif
    __syncthreads();

    const int wv   = tid >> 5;
    const int lane = tid & 31;

    // ---- Step 1: scores[t] = sum_h tile[h][t] * dlds[h] via fp32 WMMA ----
    // Waves 0..3 each produce 16 scores. A(16x4): M = 16 t-columns, K = 4 h's.
    // B(4x16): dh[k] broadcast across all N -> D rows replicate score along N.
    if (wv < 4) {
        v8f c = {};
        const int col = wv * 16 + (lane & 15);  // t-column for this lane's M
        const int ks  = (lane >> 4) << 1;       // lanes 0-15: K 0,1; lanes 16-31: K 2,3
        for (int k0 = 0; k0 < H; k0 += 4) {
            const int kk = k0 + ks;
            v2f a, bb;
            a[0]  = tile[kk * ROWSTRIDE + col];
            a[1]  = tile[(kk + 1) * ROWSTRIDE + col];
            bb[0] = dlds[kk];
            bb[1] = dlds[kk + 1];
            c = __builtin_amdgcn_wmma_f32_16x16x4_f32(
                    false, a, false, bb, (short)0, c, false, false);
        }
        // D layout: VGPR j -> M=j (lanes 0-15), M=j+8 (lanes 16-31); N = lane.
        if (lane == 0) {
            #pragma unroll
            for (int j = 0; j < 8; ++j) sc[wv * 16 + j] = c[j];
        } else if (lane == 16) {
            #pragma unroll
            for (int j = 0; j < 8; ++j) sc[wv * 16 + 8 + j] = c[j];
        }
    }
    __syncthreads();

    // ---- Step 2: per t-column softmax over h, accumulate sum over t ----
    float* myacc = accw + wv * H;  // private per-wave accumulator (deterministic)
    for (int it = 0; it < TT / NWAVES; ++it) {
        const int tl = wv * (TT / NWAVES) + it;
        const float s = sc[tl];
        float x[H / 32];
        float m = -3.402823466e38f;
        #pragma unroll
        for (int i = 0; i < H / 32; ++i) {
            x[i] = tile[(lane + 32 * i) * ROWSTRIDE + tl] * s; // bank-conflict-free
            m = fmaxf(m, x[i]);
        }
        #pragma unroll
        for (int off = 16; off >= 1; off >>= 1)
            m = fmaxf(m, __shfl_xor(m, off, 32));
        float sum = 0.0f;
        #pragma unroll
        for (int i = 0; i < H / 32; ++i) {
            const float e = __expf(x[i] - m);
            x[i] = e;
            sum += e;
        }
        #pragma unroll
        for (int off = 16; off >= 1; off >>= 1)
            sum += __shfl_xor(sum, off, 32);
        const float inv = 1.0f / sum;
        #pragma unroll
        for (int i = 0; i < H / 32; ++i)
            myacc[lane + 32 * i] += x[i] * inv;
    }
    __syncthreads();

    // Reduce the 8 per-wave accumulators, emit this tile's partial [H].
    for (int h = tid; h < H; h += NTHREADS) {
        float v = 0.0f;
        #pragma unroll
        for (int w = 0; w < NWAVES; ++w) v += accw[w * H + h];
        partial[(size_t)blockIdx.x * H + h] = v;
    }
}

__global__ __launch_bounds__(256)
void attn_reduce_kernel(const float* __restrict__ partial,
                        float* __restrict__ out)
{
    const int idx = blockIdx.x * 256 + threadIdx.x;  // 0 .. B*H-1
    const int b = idx >> 10;
    const int h = idx & (H - 1);
    float v = 0.0f;
    #pragma unroll
    for (int t = 0; t < NTILE; ++t)
        v += partial[(size_t)(b * NTILE + t) * H + h];
    out[idx] = v;  // out[b*H + h], matches [B, H] row-major
}

extern "C" void kernel_launch(void* const* d_in, const int* in_sizes, int n_in,
                              void* d_out, int out_size, void* d_ws, size_t ws_size,
                              hipStream_t stream) {
    const float* dh  = (const float*)d_in[0];   // [B, H] fp32
    const float* enc = (const float*)d_in[1];   // [H, B, T] fp32
    float* out = (float*)d_out;                 // [B, H] fp32
    float* partial = (float*)d_ws;              // needs B*NTILE*H*4 = 4 MB scratch

    attn_tile_kernel<<<dim3(B * NTILE), dim3(NTHREADS), SMEM_BYTES, stream>>>(dh, enc, partial);
    attn_reduce_kernel<<<dim3((B * H) / 256), dim3(256), 0, stream>>>(partial, out);
}